// AttentionLayer_77524159693050
// MI455X (gfx1250) — compile-verified
//
#include <hip/hip_runtime.h>

typedef __attribute__((ext_vector_type(2))) float v2f;
typedef __attribute__((ext_vector_type(4))) float v4f;
typedef __attribute__((ext_vector_type(8))) float v8f;

#define NO_   20000
#define KC_   11
#define DIM_  256
#define OUT_  128
#define K_    10

// ---------------------------------------------------------------------------
// Kernel 0: v[i] = sum_j W[i,j] * A[OUT+j]   (i in 0..511), zero accumulators.
// ---------------------------------------------------------------------------
__global__ __launch_bounds__(512) void prep_kernel(const float* __restrict__ W,
                                                   const float* __restrict__ A,
                                                   float* __restrict__ ws) {
    const int i = threadIdx.x;
    const float* wr = W + (size_t)i * OUT_;
    const float* a2 = A + OUT_;
    float s = 0.f;
    #pragma unroll 8
    for (int j = 0; j < OUT_; ++j) s = fmaf(wr[j], a2[j], s);
    ws[i] = s;
    if (i == 0) { ws[512] = 0.f; ws[513] = 0.f; }
}

// ---------------------------------------------------------------------------
// Kernel 1: fused score (WMMA f32 16x16x4) + drop-argmin + gather/copy.
// One wave (32 lanes) per n; 8 waves per block; grid = NO/8 = 2500.
// Output stores are non-temporal so the 205 MB write stream does not evict
// the 100 MB gather table from the 192 MB L2.
// ---------------------------------------------------------------------------
__global__ __launch_bounds__(256) void
attn_select_kernel(const float* __restrict__ Cand,    // [NO,KC,DIM]
                   const float* __restrict__ ndist,   // [NO,KC]
                   const int*   __restrict__ nind,    // [NO,KC]
                   const float* __restrict__ train,   // [NT,DIM]
                   const float* __restrict__ ws,      // v at ws[0..511]
                   float* __restrict__ out,           // full output buffer
                   float* __restrict__ accAB)         // 2 global accumulators
{
    __shared__ float vlds[2 * DIM_];   // the 512-vector v
    __shared__ float sc[8][16];        // per-wave scores

    const int tid  = threadIdx.x;
    const int wave = tid >> 5;
    const int lane = tid & 31;

    vlds[tid]       = ws[tid];
    vlds[tid + 256] = ws[tid + 256];
    __syncthreads();

    const int n    = blockIdx.x * 8 + wave;
    const int base = n * KC_;

    // --- WMMA score phase: 16-row tile, row M = min(M,10); A/B are f32x2/lane
    const int  mrow = (lane & 15);
    const int  rowk = mrow < (KC_ - 1) ? mrow : (KC_ - 1);
    const int  off  = (lane >> 4) << 1;           // 0 for lanes 0-15, 2 for 16-31
    const float* crow = Cand  + (size_t)(base + rowk) * DIM_;
    const float* trow = train + (size_t)nind[base + rowk] * DIM_;

    v8f acc0 = {};
    v8f acc1 = {};
    #pragma unroll 8
    for (int kb = 0; kb < DIM_; kb += 8) {
        v2f a0 = *(const v2f*)(crow + kb + off);
        v2f b0 = *(const v2f*)(&vlds[kb + off]);
        acc0 = __builtin_amdgcn_wmma_f32_16x16x4_f32(false, a0, false, b0,
                                                     (short)0, acc0, false, false);
        v2f a1 = *(const v2f*)(crow + kb + 4 + off);
        v2f b1 = *(const v2f*)(&vlds[kb + 4 + off]);
        acc1 = __builtin_amdgcn_wmma_f32_16x16x4_f32(false, a1, false, b1,
                                                     (short)0, acc1, false, false);
    }
    #pragma unroll 8
    for (int kb = 0; kb < DIM_; kb += 8) {
        v2f a0 = *(const v2f*)(trow + kb + off);
        v2f b0 = *(const v2f*)(&vlds[DIM_ + kb + off]);
        acc0 = __builtin_amdgcn_wmma_f32_16x16x4_f32(false, a0, false, b0,
                                                     (short)0, acc0, false, false);
        v2f a1 = *(const v2f*)(trow + kb + 4 + off);
        v2f b1 = *(const v2f*)(&vlds[DIM_ + kb + 4 + off]);
        acc1 = __builtin_amdgcn_wmma_f32_16x16x4_f32(false, a1, false, b1,
                                                     (short)0, acc1, false, false);
    }
    v8f acc = acc0 + acc1;

    // D layout: VGPR p -> M=p (lanes 0-15), M=p+8 (lanes 16-31); N = lane&15.
    // Column N=0 lives in lane 0 (M=0..7) and lane 16 (M=8..15).
    if (lane == 0) {
        #pragma unroll
        for (int i = 0; i < 8; ++i) sc[wave][i] = acc[i];
    }
    if (lane == 16) {
        #pragma unroll
        for (int i = 0; i < 3; ++i) sc[wave][8 + i] = acc[i];
    }
    __syncthreads();

    // --- argmin with tie -> later index (matches top_k keeping lower indices)
    float mn = sc[wave][0];
    int   m  = 0;
    #pragma unroll
    for (int k2 = 1; k2 < KC_; ++k2) {
        float s = sc[wave][k2];
        if (s <= mn) { mn = s; m = k2; }
    }

    // --- copy / gather phase: each lane owns 8 consecutive floats of a row
    const long long CANDOUT = (long long)NO_ * K_ * DIM_;   // 51,200,000
    const int d = lane * 8;
    float apart = 0.f;
    #pragma unroll
    for (int j = 0; j < K_; ++j) {
        const int src = j + (j >= m ? 1 : 0);
        const float* cr = Cand  + (size_t)(base + src) * DIM_;
        const float* tr = train + (size_t)nind[base + src] * DIM_;
        float* orow = out + (size_t)(n * K_ + j) * DIM_;
        v4f c0 = *(const v4f*)(cr + d);
        v4f c1 = *(const v4f*)(cr + d + 4);
        __builtin_nontemporal_store(c0, (v4f*)(orow + d));
        __builtin_nontemporal_store(c1, (v4f*)(orow + d + 4));
        v4f t0 = *(const v4f*)(tr + d);
        v4f t1 = *(const v4f*)(tr + d + 4);
        apart += (t0.x + t0.y + t0.z + t0.w) + (t1.x + t1.y + t1.z + t1.w);
    }
    float bpart;
    {
        const float* tr = train + (size_t)nind[base + m] * DIM_;
        v4f t0 = *(const v4f*)(tr + d);
        v4f t1 = *(const v4f*)(tr + d + 4);
        bpart = (t0.x + t0.y + t0.z + t0.w) + (t1.x + t1.y + t1.z + t1.w);
    }

    if (lane < K_) {
        const int src = lane + (lane >= m ? 1 : 0);
        __builtin_nontemporal_store(ndist[base + src],
                                    out + CANDOUT + (size_t)n * K_ + lane);
        __builtin_nontemporal_store((float)nind[base + src],
                                    out + CANDOUT + (size_t)NO_ * K_ +
                                        (size_t)n * K_ + lane);
    }

    // wave32 reduction, then one atomic per wave
    #pragma unroll
    for (int o = 16; o > 0; o >>= 1) {
        apart += __shfl_down(apart, o);
        bpart += __shfl_down(bpart, o);
    }
    if (lane == 0) {
        atomicAdd(&accAB[0], apart);
        atomicAdd(&accAB[1], bpart);
    }
}

// ---------------------------------------------------------------------------
// Kernel 2: finalize scalar outputs.
// ---------------------------------------------------------------------------
__global__ void fin_kernel(const float* __restrict__ accAB,
                           float* __restrict__ out) {
    const long long CANDOUT = (long long)NO_ * K_ * DIM_;
    out[CANDOUT + 2LL * NO_ * K_]     = accAB[0] / (float)(NO_ * K_);
    out[CANDOUT + 2LL * NO_ * K_ + 1] = accAB[1] / (float)NO_;
}

extern "C" void kernel_launch(void* const* d_in, const int* in_sizes, int n_in,
                              void* d_out, int out_size, void* d_ws, size_t ws_size,
                              hipStream_t stream) {
    // setup_inputs order: X, Candidate, neigh_dist, neigh_ind, data_m_train,
    //                     data_m_batch, test, W, A
    const float* Cand  = (const float*)d_in[1];
    const float* ndist = (const float*)d_in[2];
    const int*   nind  = (const int*)  d_in[3];
    const float* train = (const float*)d_in[4];
    const float* W     = (const float*)d_in[7];
    const float* A     = (const float*)d_in[8];
    float* out = (float*)d_out;
    float* ws  = (float*)d_ws;

    hipLaunchKernelGGL(prep_kernel, dim3(1), dim3(512), 0, stream, W, A, ws);
    hipLaunchKernelGGL(attn_select_kernel, dim3(NO_ / 8), dim3(256), 0, stream,
                       Cand, ndist, nind, train, ws, out, ws + 512);
    hipLaunchKernelGGL(fin_kernel, dim3(1), dim3(1), 0, stream, ws + 512, out);
}